// FCOSDecoder_54451595378771
// MI455X (gfx1250) — compile-verified
//
#include <hip/hip_runtime.h>
#include <math.h>

#define BATCH 16
#define NUM_CLASSES 80
#define NTOT 17064           // 12800+3200+800+208+56
#define NCP 3072             // padded candidate count (3064 -> 3072)
#define NC 3064
#define NWORDS 96            // 3072/32 bitset words per row
#define TOPN 1000
#define MIN_SCORE_C 0.01f
#define NMS_THR_C 0.6f
#define MAX_DET_C 100
#define NTILES 192           // 3072/16
#define TRI_TILES 18528      // 192*193/2
#define OUT_TOTAL 137600     // 16*100*(1+1+4+80)

typedef __attribute__((ext_vector_type(2))) float v2f;
typedef __attribute__((ext_vector_type(8))) float v8f;

__device__ __forceinline__ float sig_(float x) { return 1.0f / (1.0f + __expf(-x)); }

__device__ __forceinline__ unsigned ldsOff_(const void* p) {
  // generic (flat) shared pointer: addr[31:0] is the LDS byte address
  return (unsigned)(size_t)p;
}

// async copy: 16 bytes global -> LDS per active lane, tracked by ASYNCcnt
__device__ __forceinline__ void asyncLds16_(unsigned ldsAddr, const void* gaddr) {
  asm volatile("global_load_async_to_lds_b128 %0, %1, off"
               :: "v"(ldsAddr), "v"(gaddr) : "memory");
}
__device__ __forceinline__ void waitAsync_(int n) {
  if (n == 0) asm volatile("s_wait_asynccnt 0x0" ::: "memory");
  else        asm volatile("s_wait_asynccnt 0x1" ::: "memory");
}

__device__ __forceinline__ int levelOf_(int a, int& off) {
  if (a < 12800) { off = a;          return 0; }
  if (a < 16000) { off = a - 12800;  return 1; }
  if (a < 16800) { off = a - 16000;  return 2; }
  if (a < 17008) { off = a - 16800;  return 3; }
  off = a - 17008; return 4;
}
__device__ __forceinline__ int levelCnt_(int lvl) {
  const int c[5] = {12800, 3200, 800, 208, 56};
  return c[lvl];
}

// ---------------------------------------------------------------- init
__global__ void k_init(unsigned* suppMask, float* out) {
  size_t id = (size_t)blockIdx.x * blockDim.x + threadIdx.x;
  const size_t MW = (size_t)BATCH * NCP * NWORDS;
  if (id < MW) suppMask[id] = 0u;
  if (id < OUT_TOTAL) out[id] = -1.0f;
}

// ---------------------------------------------------------------- fused scores
__global__ void k_scores(const float* c0, const float* c1, const float* c2,
                         const float* c3, const float* c4,
                         const float* t0, const float* t1, const float* t2,
                         const float* t3, const float* t4,
                         float* scoreAll, int* clsAll) {
  int id = blockIdx.x * blockDim.x + threadIdx.x;
  if (id >= BATCH * NTOT) return;
  int img = id / NTOT, a = id - img * NTOT;
  int off, lvl = levelOf_(a, off);
  const float* cls; const float* ctr;
  switch (lvl) {
    case 0: cls = c0; ctr = t0; break;
    case 1: cls = c1; ctr = t1; break;
    case 2: cls = c2; ctr = t2; break;
    case 3: cls = c3; ctr = t3; break;
    default: cls = c4; ctr = t4; break;
  }
  int cnt = levelCnt_(lvl);
  const float4* p = (const float4*)(cls + ((size_t)img * cnt + off) * NUM_CLASSES);
  float best = -3.4e38f; int bi = 0;
  #pragma unroll 4
  for (int j = 0; j < NUM_CLASSES / 4; ++j) {
    float4 v = p[j];
    if (v.x > best) { best = v.x; bi = 4 * j + 0; }
    if (v.y > best) { best = v.y; bi = 4 * j + 1; }
    if (v.z > best) { best = v.z; bi = 4 * j + 2; }
    if (v.w > best) { best = v.w; bi = 4 * j + 3; }
  }
  float cv = ctr[(size_t)img * cnt + off];
  scoreAll[id] = sqrtf(sig_(best) * sig_(cv));
  clsAll[id] = bi;
}

// ---------------------------------------------------------------- bitonic (desc) in LDS
__device__ void bitonic_desc_(float* key, int* val, int SZ) {
  for (int k = 2; k <= SZ; k <<= 1) {
    for (int j = k >> 1; j > 0; j >>= 1) {
      __syncthreads();
      for (int e = threadIdx.x; e < SZ; e += blockDim.x) {
        int ixj = e ^ j;
        if (ixj > e) {
          float a = key[e], b = key[ixj];
          bool descBlk = ((e & k) == 0);
          bool sw = descBlk ? (a < b) : (a > b);
          if (sw) {
            key[e] = b; key[ixj] = a;
            int tv = val[e]; val[e] = val[ixj]; val[ixj] = tv;
          }
        }
      }
    }
  }
  __syncthreads();
}

// ---------------------------------------------------------------- top-1000 for levels 0,1
__global__ void k_topk(const float* scoreAll, const int* clsAll,
                       float* candScore, int* candClass, int* candAnchor) {
  extern __shared__ unsigned char smem[];
  int img = blockIdx.x >> 1;
  int lvl = blockIdx.x & 1;
  int n    = lvl ? 3200 : 12800;
  int SZ   = lvl ? 4096 : 16384;
  int base = lvl ? 12800 : 0;
  float* key = (float*)smem;
  int*   val = (int*)(key + SZ);
  // stage contiguous score block straight into LDS via async-to-LDS (16B/lane)
  const float* src = scoreAll + (size_t)img * NTOT + base;
  for (int e = threadIdx.x * 4; e < n; e += blockDim.x * 4)
    asyncLds16_(ldsOff_(&key[e]), src + e);
  for (int e = threadIdx.x; e < SZ; e += blockDim.x) {
    if (e >= n) key[e] = -1e30f;
    val[e] = (e < n) ? base + e : -1;
  }
  waitAsync_(0);
  __syncthreads();
  bitonic_desc_(key, val, SZ);
  int cbase = lvl ? TOPN : 0;
  for (int e = threadIdx.x; e < TOPN; e += blockDim.x) {
    size_t o = (size_t)img * NCP + cbase + e;
    int a = val[e];
    candScore[o]  = key[e];
    candClass[o]  = clsAll[(size_t)img * NTOT + a];
    candAnchor[o] = a;
  }
}

// ---------------------------------------------------------------- pass-through small levels
__global__ void k_pass(const float* scoreAll, const int* clsAll,
                       float* candScore, int* candClass, int* candAnchor) {
  int id = blockIdx.x * blockDim.x + threadIdx.x;
  const int per = NCP - 2000;  // 1072
  if (id >= BATCH * per) return;
  int img = id / per, r = id - img * per;
  size_t o = (size_t)img * NCP + 2000 + r;
  if (r < 1064) {
    int a = 16000 + r;
    candScore[o]  = scoreAll[(size_t)img * NTOT + a];
    candClass[o]  = clsAll[(size_t)img * NTOT + a];
    candAnchor[o] = a;
  } else {
    candScore[o] = -1.0f; candClass[o] = -1; candAnchor[o] = -1;
  }
}

// ---------------------------------------------------------------- box decode
__global__ void k_boxes(const float* r0, const float* r1, const float* r2,
                        const float* r3, const float* r4,
                        const float* p0, const float* p1, const float* p2,
                        const float* p3, const float* p4,
                        const int* candAnchor, float* candBox) {
  int id = blockIdx.x * blockDim.x + threadIdx.x;
  if (id >= BATCH * NCP) return;
  int img = id / NCP;
  int a = candAnchor[id];
  float x1 = 0.f, y1 = 0.f, x2 = 0.f, y2 = 0.f;
  if (a >= 0) {
    int off, lvl = levelOf_(a, off);
    const float* rg; const float* ps;
    switch (lvl) {
      case 0: rg = r0; ps = p0; break;
      case 1: rg = r1; ps = p1; break;
      case 2: rg = r2; ps = p2; break;
      case 3: rg = r3; ps = p3; break;
      default: rg = r4; ps = p4; break;
    }
    int cnt = levelCnt_(lvl);
    size_t rb = ((size_t)img * cnt + off) * 4;
    size_t pb = ((size_t)img * cnt + off) * 2;
    float e0 = __expf(rg[rb + 0]), e1 = __expf(rg[rb + 1]);
    float e2 = __expf(rg[rb + 2]), e3 = __expf(rg[rb + 3]);
    float px = ps[pb + 0], py = ps[pb + 1];
    x1 = fmaxf(truncf(px - e0), 0.0f);
    y1 = fmaxf(truncf(py - e1), 0.0f);
    x2 = fminf(truncf(px + e2), 1023.0f);
    y2 = fminf(truncf(py + e3), 799.0f);
  }
  candBox[(size_t)id * 4 + 0] = x1;
  candBox[(size_t)id * 4 + 1] = y1;
  candBox[(size_t)id * 4 + 2] = x2;
  candBox[(size_t)id * 4 + 3] = y2;
}

// ---------------------------------------------------------------- global sort per image + keep init
__global__ void k_sortcand(const float* candScore, const int* candClass,
                           const int* candAnchor, const float* candBox,
                           float* sScore, int* sClass, int* sAnchor, float* sBox,
                           unsigned* keepInit) {
  extern __shared__ unsigned char smem[];
  const int SZ = 4096;
  int img = blockIdx.x;
  float* key = (float*)smem;
  int*   val = (int*)(key + SZ);
  // async-stage raw candidate scores into LDS, then transform in place
  const float* src = candScore + (size_t)img * NCP;
  for (int e = threadIdx.x * 4; e < NCP; e += blockDim.x * 4)
    asyncLds16_(ldsOff_(&key[e]), src + e);
  waitAsync_(0);
  __syncthreads();
  for (int e = threadIdx.x; e < SZ; e += blockDim.x) {
    if (e < NCP) {
      float s = key[e];
      key[e] = (s > MIN_SCORE_C) ? s : -1e30f;   // invalid to the back
      val[e] = e;
    } else { key[e] = -2e30f; val[e] = -1; }
  }
  bitonic_desc_(key, val, SZ);
  for (int e = threadIdx.x; e < NCP; e += blockDim.x) {
    int s = val[e];
    size_t so = (size_t)img * NCP + s;
    size_t de = (size_t)img * NCP + e;
    sScore[de]  = candScore[so];
    sClass[de]  = candClass[so];
    sAnchor[de] = candAnchor[so];
    sBox[de * 4 + 0] = candBox[so * 4 + 0];
    sBox[de * 4 + 1] = candBox[so * 4 + 1];
    sBox[de * 4 + 2] = candBox[so * 4 + 2];
    sBox[de * 4 + 3] = candBox[so * 4 + 3];
  }
  __syncthreads();
  // wave32 ballot -> initial keep bitset (valid candidates)
  for (int pass = 0; pass < SZ / 1024; ++pass) {
    int e = threadIdx.x + pass * 1024;
    bool valid = (e < NCP) && (key[e] > MIN_SCORE_C);
    unsigned m = (unsigned)__ballot(valid);
    if (e < NCP && (threadIdx.x & 31) == 0) keepInit[img * NWORDS + (e >> 5)] = m;
  }
}

// ---------------------------------------------------------------- suppression bitmask via WMMA + ballot
__device__ __forceinline__ int triF_(int ti) { return ti * NTILES - (ti * (ti - 1)) / 2; }

__global__ void k_mask(const float* sBox, unsigned* suppMask) {
  int wave = blockIdx.x * (blockDim.x >> 5) + (threadIdx.x >> 5);
  int lane = threadIdx.x & 31;
  if (wave >= BATCH * TRI_TILES) return;
  int img = wave / TRI_TILES;
  int t   = wave - img * TRI_TILES;
  // decode upper-triangular (ti, tj), ti <= tj
  int ti = (int)((385.0f - sqrtf(148225.0f - 8.0f * (float)t)) * 0.5f);
  if (ti < 0) ti = 0; if (ti > NTILES - 1) ti = NTILES - 1;
  while (ti > 0 && triF_(ti) > t) --ti;
  while (triF_(ti + 1) <= t) ++ti;
  int tj = ti + (t - triF_(ti));

  int N = lane & 15, half = lane >> 4;
  size_t cb = ((size_t)img * NCP + tj * 16 + N) * 4;
  float cx1 = sBox[cb + 0], cy1 = sBox[cb + 1], cx2 = sBox[cb + 2], cy2 = sBox[cb + 3];
  float cArea = (cx2 - cx1) * (cy2 - cy1);
  size_t rb = ((size_t)img * NCP + ti * 16 + N) * 4;  // lanes 0..15 carry row boxes
  float rx1 = sBox[rb + 0], ry1 = sBox[rb + 1], rx2 = sBox[rb + 2], ry2 = sBox[rb + 3];
  float rArea = (rx2 - rx1) * (ry2 - ry1);

  // D[M,N] = areaR[M] + areaC[N]  via A=[areaR,1,0,0] (16x4), B=[1;areaC;0;0] (4x16)
  v2f A, B; v8f C = {};
  A.x = half ? 0.0f : rArea;  A.y = half ? 0.0f : 1.0f;   // K=0 / K=1 (K=2,3 zero)
  B.x = half ? 0.0f : 1.0f;   B.y = half ? 0.0f : cArea;  // row0 ones, row1 areaC
  v8f D = __builtin_amdgcn_wmma_f32_16x16x4_f32(false, A, false, B, (short)0, C, false, false);

  unsigned ball[8];
  #pragma unroll
  for (int r = 0; r < 8; ++r) {
    int M = r + (half << 3);                 // C layout: vgpr r, lanes>=16 -> M = r+8
    float bx1 = __shfl(rx1, M, 32);
    float by1 = __shfl(ry1, M, 32);
    float bx2 = __shfl(rx2, M, 32);
    float by2 = __shfl(ry2, M, 32);
    float iw = fmaxf(fminf(bx2, cx2) - fmaxf(bx1, cx1), 0.0f);
    float ih = fmaxf(fminf(by2, cy2) - fmaxf(by1, cy1), 0.0f);
    float inter = iw * ih;
    float iou = inter / (D[r] - inter);
    int gi = ti * 16 + M, gj = tj * 16 + N;
    bool pred = (iou > NMS_THR_C) && (gj > gi);
    ball[r] = (unsigned)__ballot(pred);      // low16: row r, high16: row r+8
  }
  if (lane == 0) {
    int word = (tj * 16) >> 5, shift = (tj * 16) & 31;
    unsigned* mbase = suppMask + (size_t)img * NCP * NWORDS;
    #pragma unroll
    for (int r = 0; r < 8; ++r) {
      unsigned lo = ball[r] & 0xFFFFu, hi = ball[r] >> 16;
      if (lo) atomicOr(&mbase[(size_t)(ti * 16 + r) * NWORDS + word], lo << shift);
      if (hi) atomicOr(&mbase[(size_t)(ti * 16 + r + 8) * NWORDS + word], hi << shift);
    }
  }
}

// ---------------------------------------------------------------- sequential bitset NMS
// Double-buffered: wave 0 async-streams mask row i+1 into LDS while the block
// ANDs row i out of the keep bitset.  ASYNCcnt hides the entire row latency.
__global__ void k_nms(const unsigned* keepInit, const unsigned* suppMask,
                      unsigned* keepFinal) {
  int img = blockIdx.x, tid = threadIdx.x;
  __shared__ unsigned keep[NWORDS];
  __shared__ __align__(16) unsigned rowbuf[2][NWORDS];
  if (tid < NWORDS) keep[tid] = keepInit[img * NWORDS + tid];
  const unsigned* base = suppMask + (size_t)img * NCP * NWORDS;
  // prime: row 0 -> buf 0 (24 lanes x 16B = 384B row)
  if (tid < 24)
    asyncLds16_(ldsOff_(&rowbuf[0][tid * 4]), base + tid * 4);
  __syncthreads();
  for (int i = 0; i < NC; ++i) {
    int buf = i & 1;
    if (tid < 24 && (i + 1) < NC)
      asyncLds16_(ldsOff_(&rowbuf[buf ^ 1][tid * 4]),
                  base + (size_t)(i + 1) * NWORDS + tid * 4);
    waitAsync_((i + 1) < NC ? 1 : 0);   // row i resident (in-order completion)
    unsigned w = keep[i >> 5];
    bool alive = (w >> (i & 31)) & 1u;
    __syncthreads();
    if (alive && tid < NWORDS) keep[tid] &= ~rowbuf[buf][tid];
    __syncthreads();
  }
  if (tid < NWORDS) keepFinal[img * NWORDS + tid] = keep[tid];
}

// ---------------------------------------------------------------- output writer
__global__ void k_output(const float* c0, const float* c1, const float* c2,
                         const float* c3, const float* c4,
                         const float* sScore, const int* sClass,
                         const int* sAnchor, const float* sBox,
                         const unsigned* keepFinal, float* out) {
  int img = blockIdx.x, tid = threadIdx.x;
  __shared__ unsigned kw[NWORDS];
  __shared__ int pfx[NWORDS];
  if (tid < NWORDS) kw[tid] = keepFinal[img * NWORDS + tid];
  __syncthreads();
  if (tid == 0) {
    int run = 0;
    for (int w = 0; w < NWORDS; ++w) { pfx[w] = run; run += __popc(kw[w]); }
  }
  __syncthreads();
  float* out_s  = out;
  float* out_c  = out + BATCH * MAX_DET_C;
  float* out_b  = out + 2 * BATCH * MAX_DET_C;
  float* out_cp = out + 2 * BATCH * MAX_DET_C + BATCH * MAX_DET_C * 4;
  for (int e = tid; e < NC; e += blockDim.x) {
    unsigned w = kw[e >> 5];
    if (!((w >> (e & 31)) & 1u)) continue;
    int rank = pfx[e >> 5] + __popc(w & ((1u << (e & 31)) - 1u));
    if (rank >= MAX_DET_C) continue;
    size_t so = (size_t)img * NCP + e;
    int od = img * MAX_DET_C + rank;
    out_s[od] = sScore[so];
    out_c[od] = (float)sClass[so];
    out_b[(size_t)od * 4 + 0] = sBox[so * 4 + 0];
    out_b[(size_t)od * 4 + 1] = sBox[so * 4 + 1];
    out_b[(size_t)od * 4 + 2] = sBox[so * 4 + 2];
    out_b[(size_t)od * 4 + 3] = sBox[so * 4 + 3];
    int a = sAnchor[so];
    int off, lvl = levelOf_(a, off);
    const float* cls;
    switch (lvl) {
      case 0: cls = c0; break; case 1: cls = c1; break; case 2: cls = c2; break;
      case 3: cls = c3; break; default: cls = c4; break;
    }
    const float* row = cls + ((size_t)img * levelCnt_(lvl) + off) * NUM_CLASSES;
    for (int k = 0; k < NUM_CLASSES; ++k)
      out_cp[(size_t)od * NUM_CLASSES + k] = sig_(row[k]);
  }
}

// ---------------------------------------------------------------- launch
extern "C" void kernel_launch(void* const* d_in, const int* in_sizes, int n_in,
                              void* d_out, int out_size, void* d_ws, size_t ws_size,
                              hipStream_t stream) {
  const float *cls[5], *reg[5], *ctr[5], *pos[5];
  for (int i = 0; i < 5; ++i) {
    cls[i] = (const float*)d_in[4 * i + 0];
    reg[i] = (const float*)d_in[4 * i + 1];
    ctr[i] = (const float*)d_in[4 * i + 2];
    pos[i] = (const float*)d_in[4 * i + 3];
  }
  char* w = (char*)d_ws;
  auto take = [&](size_t bytes) -> char* {
    char* r = w; w += (bytes + 255) & ~(size_t)255; return r;
  };
  float*    scoreAll   = (float*)take((size_t)BATCH * NTOT * 4);
  int*      clsAll     = (int*)  take((size_t)BATCH * NTOT * 4);
  float*    candScore  = (float*)take((size_t)BATCH * NCP * 4);
  int*      candClass  = (int*)  take((size_t)BATCH * NCP * 4);
  int*      candAnchor = (int*)  take((size_t)BATCH * NCP * 4);
  float*    candBox    = (float*)take((size_t)BATCH * NCP * 16);
  float*    sScore     = (float*)take((size_t)BATCH * NCP * 4);
  int*      sClass     = (int*)  take((size_t)BATCH * NCP * 4);
  int*      sAnchor    = (int*)  take((size_t)BATCH * NCP * 4);
  float*    sBox       = (float*)take((size_t)BATCH * NCP * 16);
  unsigned* keepInit   = (unsigned*)take((size_t)BATCH * NWORDS * 4);
  unsigned* keepFinal  = (unsigned*)take((size_t)BATCH * NWORDS * 4);
  unsigned* suppMask   = (unsigned*)take((size_t)BATCH * NCP * NWORDS * 4);

  float* out = (float*)d_out;

  {
    size_t n = (size_t)BATCH * NCP * NWORDS;
    int blocks = (int)((n + 255) / 256);
    k_init<<<blocks, 256, 0, stream>>>(suppMask, out);
  }
  {
    int n = BATCH * NTOT;
    k_scores<<<(n + 255) / 256, 256, 0, stream>>>(
        cls[0], cls[1], cls[2], cls[3], cls[4],
        ctr[0], ctr[1], ctr[2], ctr[3], ctr[4], scoreAll, clsAll);
  }
  k_topk<<<BATCH * 2, 1024, 16384 * 8, stream>>>(scoreAll, clsAll,
                                                 candScore, candClass, candAnchor);
  {
    int n = BATCH * (NCP - 2000);
    k_pass<<<(n + 255) / 256, 256, 0, stream>>>(scoreAll, clsAll,
                                                candScore, candClass, candAnchor);
  }
  {
    int n = BATCH * NCP;
    k_boxes<<<(n + 255) / 256, 256, 0, stream>>>(
        reg[0], reg[1], reg[2], reg[3], reg[4],
        pos[0], pos[1], pos[2], pos[3], pos[4], candAnchor, candBox);
  }
  k_sortcand<<<BATCH, 1024, 4096 * 8, stream>>>(candScore, candClass, candAnchor,
                                                candBox, sScore, sClass, sAnchor,
                                                sBox, keepInit);
  {
    int waves = BATCH * TRI_TILES;
    int wavesPerBlock = 4;  // 128 threads
    int blocks = (waves + wavesPerBlock - 1) / wavesPerBlock;
    k_mask<<<blocks, 128, 0, stream>>>(sBox, suppMask);
  }
  k_nms<<<BATCH, 128, 0, stream>>>(keepInit, suppMask, keepFinal);
  k_output<<<BATCH, 128, 0, stream>>>(cls[0], cls[1], cls[2], cls[3], cls[4],
                                      sScore, sClass, sAnchor, sBox, keepFinal, out);
}